// RangedCoulombPool_32375463477476
// MI455X (gfx1250) — compile-verified
//
#include <hip/hip_runtime.h>
#include <hip/hip_bf16.h>

// ---------------------------------------------------------------------------
// out[b,n,k] = S_k * sum_m sinc(S_k * ||R_n - r_m||) * f[b,m]
//            = sum_m sin(S_k * d) * f[b,m] / d          (S_k prefactor cancels)
//
// MI455X reasoning: inputs ~250KB, output 512KB -> bandwidth irrelevant at
// 23.3 TB/s. 537M v_sin_f32 evaluations dominate (transcendental roofline).
// We use V_WMMA_F32_16X16X4_F32 to produce 16x16 (n x m) tiles of
// (|r_m|^2 - 2 R_n . r_m) in a single matrix op (A = [-2x,-2y,-2z,1],
// B = [x; y; z; |r|^2]), then d^2 = tile + |R_n|^2. This strips the per-pair
// distance VALU work down to ~3 non-trans ops per sin, i.e. sits on the
// v_sin throughput ceiling. K=16 is split into 4 groups of 4 across blocks:
// 32 accumulator VGPRs/lane, 2048 waves, and every output element is owned
// by exactly one lane -> plain stores, bit-deterministic (no fp atomics).
// ---------------------------------------------------------------------------

typedef __attribute__((ext_vector_type(2))) float v2f;
typedef __attribute__((ext_vector_type(8))) float v8f;

#define B_      2
#define N_      4096
#define M_      4096
#define K_      16
#define KCH     4                 // k-channels per wave (4 groups via blockIdx)
#define NTILE   16
#define WAVES   8                 // waves per block (block = 256 = 8 x wave32)
#define CHUNK   256               // m points staged in LDS per iteration
#define INV2PI  0.15915494309189535f
#define LN2F    0.6931471805599453f

__global__ __launch_bounds__(256)
void sinc_field_wmma(const float* __restrict__ f,       // [B, M, 1]
                     const float* __restrict__ coords,  // [B, M, 3]
                     const float* __restrict__ outc,    // [B, N, 3]
                     const float* __restrict__ kbasis,  // [K]
                     float* __restrict__ out)           // [B, N, K]
{
    __shared__ float4 pts[CHUNK];            // staged m-points: x,y,z,f
    __shared__ float  rn2s[WAVES * NTILE];   // |R_n|^2 per wave's 16 rows

    const int tid  = threadIdx.x;
    const int w    = tid >> 5;       // wave in block
    const int lane = tid & 31;
    const int col  = lane & 15;      // tile column / row selector
    const int hi   = lane >> 4;      // half-wave: selects A/B K-components & D rows

    const int kgroup = blockIdx.x >> 6;              // 0..3  (k-range)
    const int ntgrp  = blockIdx.x & 63;              // 0..63
    const int ntile  = ntgrp * WAVES + w;            // 0..511 (global 16-row tile)
    const int b      = ntile >> 8;                   // batch (256 tiles per batch)
    const int nbase  = (ntile & 255) * NTILE;

    // --- basis constants: c[kk] = S_k / (2*pi), hardware v_sin takes revolutions
    float c[KCH];
#pragma unroll
    for (int kk = 0; kk < KCH; ++kk) {
        float kv = kbasis[kgroup * KCH + kk];        // uniform -> scalar load
        float ax = fabsf(kv);
        float S  = ax + log1pf(__expf(-2.0f * ax)) - LN2F + 1e-5f;
        c[kk] = S * INV2PI;
    }

    // --- A matrix (16x4 f32): row = lane%16; lo lanes hold K=0,1 = (-2x,-2y),
    //     hi lanes hold K=2,3 = (-2z, 1)   [ISA 7.12.2 32-bit A layout]
    v2f Av;
    {
        const int n = nbase + col;
        const float* rp = outc + ((size_t)b * N_ + n) * 3;
        float x = rp[0], y = rp[1], z = rp[2];
        if (hi == 0) rn2s[w * NTILE + col] = x * x + y * y + z * z;
        Av.x = hi ? (-2.0f * z) : (-2.0f * x);
        Av.y = hi ? 1.0f        : (-2.0f * y);
    }
    __syncthreads();

    // D-tile rows owned by this lane: v + 8*hi  (v = 0..7)
    float rn2p[8];
#pragma unroll
    for (int v = 0; v < 8; ++v)
        rn2p[v] = rn2s[w * NTILE + hi * 8 + v];

    float acc[8][KCH];
#pragma unroll
    for (int v = 0; v < 8; ++v)
#pragma unroll
        for (int kk = 0; kk < KCH; ++kk) acc[v][kk] = 0.0f;

    const float* cb = coords + (size_t)b * (M_ * 3);
    const float* fb = f + (size_t)b * M_;

    const v8f zero8 = {0.f, 0.f, 0.f, 0.f, 0.f, 0.f, 0.f, 0.f};

    for (int chunk = 0; chunk < M_ / CHUNK; ++chunk) {
        __syncthreads();
        {   // cooperative stage of 256 m-points into LDS as float4
            const int m = chunk * CHUNK + tid;
            float4 p;
            p.x = cb[m * 3 + 0];
            p.y = cb[m * 3 + 1];
            p.z = cb[m * 3 + 2];
            p.w = fb[m];
            pts[tid] = p;
        }
        __syncthreads();

#pragma unroll 1
        for (int t = 0; t < CHUNK / NTILE; ++t) {
            // this lane's m-column data (ds_load_b128), shared by its 8 rows
            const float4 q = pts[t * NTILE + col];
            const float rm2 = q.x * q.x + q.y * q.y + q.z * q.z;

            // B matrix (4x16 f32): col = lane%16; lo lanes K=0,1 = (x,y),
            // hi lanes K=2,3 = (z, |r|^2)
            v2f Bv;
            Bv.x = hi ? q.z : q.x;
            Bv.y = hi ? rm2 : q.y;

            // tD[v] = |r_m|^2 - 2 R_n . r_m   for row v+8*hi, col = lane%16
            v8f tD = __builtin_amdgcn_wmma_f32_16x16x4_f32(
                false, Av, false, Bv, (short)0, zero8, false, false);

            const float fm = q.w;
#pragma unroll
            for (int v = 0; v < 8; ++v) {
                // Gram-form cancellation can round d2 slightly negative for
                // near-coincident points -> clamp before v_rsq
                float d2   = fmaxf(tD[v] + rn2p[v], 1e-12f);
                float invd = __builtin_amdgcn_rsqf(d2);     // v_rsq_f32
                float d    = d2 * invd;
                float wgt  = fm * invd;                     // f_m / d
#pragma unroll
                for (int kk = 0; kk < KCH; ++kk) {
                    // v_mul + v_sin + v_fmac : the roofline-critical triple
                    acc[v][kk] = __builtin_fmaf(
                        __builtin_amdgcn_sinf(c[kk] * d), wgt, acc[v][kk]);
                }
            }
        }
    }

    // --- reduce over the 16 m-columns (xor-shuffle within each half-wave)
#pragma unroll
    for (int off = 8; off >= 1; off >>= 1)
#pragma unroll
        for (int v = 0; v < 8; ++v)
#pragma unroll
            for (int kk = 0; kk < KCH; ++kk)
                acc[v][kk] += __shfl_xor(acc[v][kk], off, 16);

    // --- writer lanes: lane (hi*16 + kk), kk < KCH, stores its 8 rows.
    // Each out element written by exactly one lane of one wave -> deterministic.
    if (col < KCH) {
        const int kk = col;
#pragma unroll
        for (int v = 0; v < 8; ++v) {
            const int n = nbase + hi * 8 + v;
            out[((size_t)b * N_ + n) * K_ + kgroup * KCH + kk] = acc[v][kk];
        }
    }
}

extern "C" void kernel_launch(void* const* d_in, const int* in_sizes, int n_in,
                              void* d_out, int out_size, void* d_ws, size_t ws_size,
                              hipStream_t stream) {
    (void)in_sizes; (void)n_in; (void)d_ws; (void)ws_size; (void)out_size;
    const float* f      = (const float*)d_in[0];  // [B,M,1]
    const float* coords = (const float*)d_in[1];  // [B,M,3]
    const float* outc   = (const float*)d_in[2];  // [B,N,3]
    const float* kb     = (const float*)d_in[3];  // [K]
    float* out          = (float*)d_out;          // [B,N,K]

    // 256 blocks = 64 ntile-groups x 4 k-groups; 8 waves/block = 2048 waves
    sinc_field_wmma<<<dim3(64 * (K_ / KCH), 1, 1), dim3(256, 1, 1), 0, stream>>>(
        f, coords, outc, kb, out);
}